// BaseNet_75256416960712
// MI455X (gfx1250) — compile-verified
//
#include <hip/hip_runtime.h>
#include <math.h>

typedef __attribute__((ext_vector_type(2))) float v2f;
typedef __attribute__((ext_vector_type(8))) float v8f;

#define Bb 64
#define Tt 24
#define Nn 64
#define Dd 32
#define Hh 64
#define MNb 15
#define Rr (Bb*Tt*Nn)      /* 98304 rows of the feature GEMM */
#define RH (Rr*Hh)         /* 6291456 elements of x */

// ---------------- K0: zero accumulators ----------------
__global__ void k_zero(float* __restrict__ p, int n) {
    int i = blockIdx.x * 256 + threadIdx.x;
    if (i < n) p[i] = 0.f;
}

// ---------------- K1: x = s @ W_feat + b_feat via V_WMMA_F32_16X16X4_F32 ----------------
// One wave computes a 16(M: flattened b,t,n rows) x 16(N: H cols) tile, 8 WMMAs over K=32.
__global__ __launch_bounds__(256) void k_gemm(const float* __restrict__ s,
                                              const float* __restrict__ W,
                                              const float* __restrict__ bias,
                                              float* __restrict__ x)
{
    const int lane = threadIdx.x & 31;
    const int w    = blockIdx.x * 8 + (threadIdx.x >> 5);   // global wave id, 0..24575
    const int row0 = (w >> 2) * 16;
    const int n0   = (w & 3) * 16;
    const int lh   = lane & 15;          // row (A) / col (B,D) within tile
    const int lg   = lane >> 4;          // half-wave select

    // A 16x4 f32 layout: lanes 0-15 rows, VGPR0/1 = K 0/1; lanes 16-31: K 2/3
    const float* ap = s + (size_t)(row0 + lh) * Dd + 2 * lg;
    // B 4x16 f32 layout: lanes 0-15 cols with K rows 0/1; lanes 16-31: K rows 2/3
    const float* bp = W + (size_t)(2 * lg) * Hh + n0 + lh;

    v8f c = {0.f, 0.f, 0.f, 0.f, 0.f, 0.f, 0.f, 0.f};
#pragma unroll
    for (int kk = 0; kk < 8; ++kk) {
        v2f a = *(const v2f*)(ap + 4 * kk);      // 8B aligned
        v2f b;
        b.x = bp[(size_t)(4 * kk + 0) * Hh];
        b.y = bp[(size_t)(4 * kk + 1) * Hh];
        c = __builtin_amdgcn_wmma_f32_16x16x4_f32(false, a, false, b, (short)0, c, false, false);
    }

    const float bv = bias[n0 + lh];
    // D layout: VGPR j = row j (lanes 0-15) / row j+8 (lanes 16-31), col = lh
    float* op = x + (size_t)(row0 + 8 * lg) * Hh + n0 + lh;
#pragma unroll
    for (int m = 0; m < 8; ++m)
        op[(size_t)m * Hh] = c[m] + bv;
}

// ---------------- K2: per-H-channel sum / sumsq over all rows ----------------
__global__ __launch_bounds__(256) void k_bn1_stats(const float* __restrict__ x,
                                                   float* __restrict__ sum,
                                                   float* __restrict__ sq)
{
    __shared__ float ls1[Hh], ls2[Hh];
    if (threadIdx.x < Hh) { ls1[threadIdx.x] = 0.f; ls2[threadIdx.x] = 0.f; }
    __syncthreads();
    const int gt = blockIdx.x * 256 + threadIdx.x;
    const int stride = gridDim.x * 256;             // multiple of 64 -> fixed channel per thread
    float s1 = 0.f, s2 = 0.f;
    for (int i = gt; i < RH; i += stride) { float v = x[i]; s1 += v; s2 += v * v; }
    const int ch = gt & 63;
    atomicAdd(&ls1[ch], s1);
    atomicAdd(&ls2[ch], s2);
    __syncthreads();
    if (threadIdx.x < Hh) {
        atomicAdd(&sum[threadIdx.x], ls1[threadIdx.x]);
        atomicAdd(&sq[threadIdx.x],  ls2[threadIdx.x]);
    }
}

// ---------------- K3: fold BN1 affine into effective head weights ----------------
__global__ __launch_bounds__(64) void k_bn1_fold(const float* __restrict__ sum,
                                                 const float* __restrict__ sq,
                                                 const float* __restrict__ g,
                                                 const float* __restrict__ be,
                                                 const float* __restrict__ wmu,
                                                 const float* __restrict__ bmu,
                                                 const float* __restrict__ wlv,
                                                 const float* __restrict__ blv,
                                                 float* __restrict__ weff,   // [128]
                                                 float* __restrict__ cns)    // [2]
{
    __shared__ float rm[64], rl[64];
    const int h = threadIdx.x;
    const float cnt  = (float)Rr;
    const float mean = sum[h] / cnt;
    const float var  = sq[h] / cnt - mean * mean;
    const float a = g[h] * rsqrtf(var + 1e-5f);
    const float c = be[h] - mean * a;
    weff[h]      = a * wmu[h];
    weff[64 + h] = a * wlv[h];
    rm[h] = c * wmu[h];
    rl[h] = c * wlv[h];
    __syncthreads();
    for (int off = 32; off > 0; off >>= 1) {
        if (h < off) { rm[h] += rm[h + off]; rl[h] += rl[h + off]; }
        __syncthreads();
    }
    if (h == 0) { cns[0] = bmu[0] + rm[0]; cns[1] = blv[0] + rl[0]; }
}

// ---------------- K4: y_mu/y_lv = x . w_eff per row (one row per wave32) ----------------
__global__ __launch_bounds__(256) void k_heads(const float* __restrict__ x,
                                               const float* __restrict__ weff,
                                               const float* __restrict__ cns,
                                               float* __restrict__ ymu,
                                               float* __restrict__ ylv)
{
    const int lane = threadIdx.x & 31;
    const int row  = blockIdx.x * 8 + (threadIdx.x >> 5);
    v2f xv = *(const v2f*)(x + (size_t)row * Hh + 2 * lane);
    v2f wm = *(const v2f*)(weff + 2 * lane);
    v2f wl = *(const v2f*)(weff + 64 + 2 * lane);
    float pm = xv.x * wm.x + xv.y * wm.y;
    float pl = xv.x * wl.x + xv.y * wl.y;
#pragma unroll
    for (int o = 16; o > 0; o >>= 1) {
        pm += __shfl_xor(pm, o, 32);
        pl += __shfl_xor(pl, o, 32);
    }
    if (lane == 0) { ymu[row] = pm + cns[0]; ylv[row] = pl + cns[1]; }
}

// ---------------- K5: BN2 per-N stats over (B,T,1+MN) with neighbor gather ----------------
__global__ __launch_bounds__(256) void k_bn2_stats(const float* __restrict__ ymu,
                                                   const float* __restrict__ ylv,
                                                   const int* __restrict__ knei,
                                                   float* __restrict__ st)   // [Nn*4]
{
    const int n = blockIdx.x;
    float sm = 0.f, qm = 0.f, sl = 0.f, ql = 0.f;
    for (int i = threadIdx.x; i < Bb * Tt * 16; i += 256) {
        const int b = i / (Tt * 16);
        const int r = i - b * (Tt * 16);
        const int t = r >> 4;
        const int j = r & 15;
        const int col = (j == 0) ? n : knei[(b * Nn + n) * MNb + (j - 1)];
        const int idx = (b * Tt + t) * Nn + col;
        const float vm = ymu[idx], vl = ylv[idx];
        sm += vm; qm += vm * vm; sl += vl; ql += vl * vl;
    }
    __shared__ float a0[256], a1[256], a2[256], a3[256];
    a0[threadIdx.x] = sm; a1[threadIdx.x] = qm; a2[threadIdx.x] = sl; a3[threadIdx.x] = ql;
    __syncthreads();
    for (int off = 128; off > 0; off >>= 1) {
        if (threadIdx.x < off) {
            a0[threadIdx.x] += a0[threadIdx.x + off];
            a1[threadIdx.x] += a1[threadIdx.x + off];
            a2[threadIdx.x] += a2[threadIdx.x + off];
            a3[threadIdx.x] += a3[threadIdx.x + off];
        }
        __syncthreads();
    }
    if (threadIdx.x == 0) {
        st[n * 4 + 0] = a0[0]; st[n * 4 + 1] = a1[0];
        st[n * 4 + 2] = a2[0]; st[n * 4 + 3] = a3[0];
    }
}

// ---------------- K6: final gather + BN2 + tanh/clip/exp + reparam ----------------
__global__ __launch_bounds__(256) void k_final(const float* __restrict__ ymu,
                                               const float* __restrict__ ylv,
                                               const int* __restrict__ knei,
                                               const float* __restrict__ st,
                                               const float* __restrict__ gmu,
                                               const float* __restrict__ bemu,
                                               const float* __restrict__ glv,
                                               const float* __restrict__ belv,
                                               const float* __restrict__ eps,
                                               float* __restrict__ out)
{
    const int i = blockIdx.x * 256 + threadIdx.x;      // [B,N,T,16] flat
    const int j = i & 15;
    const int t = (i >> 4) % Tt;
    const int n = (i / (16 * Tt)) % Nn;
    const int b = i / (16 * Tt * Nn);
    const int col = (j == 0) ? n : knei[(b * Nn + n) * MNb + (j - 1)];
    const int idx = (b * Tt + t) * Nn + col;
    const float pm = ymu[idx], pl = ylv[idx];
    const float cnt = (float)(Bb * Tt * 16);
    const float mm = st[n * 4 + 0] / cnt;
    const float vm = st[n * 4 + 1] / cnt - mm * mm;
    const float ml = st[n * 4 + 2] / cnt;
    const float vl = st[n * 4 + 3] / cnt - ml * ml;
    const float bnm = (pm - mm) * (gmu[n] * rsqrtf(vm + 1e-5f)) + bemu[n];
    const float bnl = (pl - ml) * (glv[n] * rsqrtf(vl + 1e-5f)) + belv[n];
    const float mu = tanhf(bnm);
    const float lv = fminf(fmaxf(bnl, -20.f), 2.f);
    out[i] = eps[i] * expf(lv) + mu;
}

extern "C" void kernel_launch(void* const* d_in, const int* in_sizes, int n_in,
                              void* d_out, int out_size, void* d_ws, size_t ws_size,
                              hipStream_t stream)
{
    (void)in_sizes; (void)n_in; (void)out_size; (void)ws_size;
    const float* s    = (const float*)d_in[0];
    const float* Wf   = (const float*)d_in[1];
    const float* bf   = (const float*)d_in[2];
    const float* ginp = (const float*)d_in[3];
    const float* beinp= (const float*)d_in[4];
    const float* wmu  = (const float*)d_in[5];
    const float* bmu  = (const float*)d_in[6];
    const float* gmu  = (const float*)d_in[7];
    const float* bemu = (const float*)d_in[8];
    const float* wlv  = (const float*)d_in[9];
    const float* blv  = (const float*)d_in[10];
    const float* glv  = (const float*)d_in[11];
    const float* belv = (const float*)d_in[12];
    const float* eps  = (const float*)d_in[13];
    const int*   knei = (const int*)d_in[14];
    float* out = (float*)d_out;

    float* ws   = (float*)d_ws;
    float* X    = ws;                  // RH floats (25.2 MB)
    float* SUM  = X + RH;              // 64
    float* SQ   = SUM + 64;            // 64
    float* WEFF = SQ + 64;             // 128
    float* CNS  = WEFF + 128;          // 4
    float* YMU  = CNS + 4;             // Rr
    float* YLV  = YMU + Rr;            // Rr
    float* ST   = YLV + Rr;            // 256

    k_zero<<<1, 256, 0, stream>>>(SUM, 128);                 // SUM + SQ contiguous
    k_gemm<<<3072, 256, 0, stream>>>(s, Wf, bf, X);          // 24576 waves, one 16x16 tile each
    k_bn1_stats<<<256, 256, 0, stream>>>(X, SUM, SQ);
    k_bn1_fold<<<1, 64, 0, stream>>>(SUM, SQ, ginp, beinp, wmu, bmu, wlv, blv, WEFF, CNS);
    k_heads<<<12288, 256, 0, stream>>>(X, WEFF, CNS, YMU, YLV);
    k_bn2_stats<<<64, 256, 0, stream>>>(YMU, YLV, knei, ST);
    k_final<<<6144, 256, 0, stream>>>(YMU, YLV, knei, ST, gmu, bemu, glv, belv, eps, out);
}